// Dir_Encoder_GCN_43301860278532
// MI455X (gfx1250) — compile-verified
//
#include <hip/hip_runtime.h>
#include <math.h>

typedef __attribute__((ext_vector_type(2))) float f32x2;
typedef __attribute__((ext_vector_type(8))) float f32x8;

#define CIN  128
#define CHID 128
#define COUT 64

// ---------------------------------------------------------------------------
// WMMA fp32 GEMM:  Out[M, NCOL] = A[M, 128] * B[128, NCOL]
// One wave per 16-column tile; B fragments preloaded in registers;
// A 16x128 strip staged in LDS (padded rows -> conflict-free b64 reads).
// Emits v_wmma_f32_16x16x4_f32 (32 per output tile).
// ---------------------------------------------------------------------------
template <int NCOL>
__global__ void __launch_bounds__(2 * NCOL)
gemm_wmma_f32(const float* __restrict__ A, const float* __restrict__ B,
              float* __restrict__ Out, int M) {
  constexpr int K   = 128;
  constexpr int LDW = K + 4;  // pad 4 floats -> 64-bank conflict-free
  __shared__ float Alds[16 * LDW];

  const int lane   = threadIdx.x & 31;
  const int wave   = threadIdx.x >> 5;          // == column tile index
  const int row_in = lane & 15;
  const int kb     = (lane >> 4) * 2;           // K sub-pair within 4-chunk
  const int col    = wave * 16 + (lane & 15);

  // Preload all B fragments for this wave's column tile (W is hot in L1/L2).
  f32x2 bf[K / 4];
#pragma unroll
  for (int ks = 0; ks < K / 4; ++ks) {
    const int kr = ks * 4 + kb;
    bf[ks].x = B[(size_t)kr * NCOL + col];
    bf[ks].y = B[(size_t)(kr + 1) * NCOL + col];
  }

  const int mtiles = (M + 15) >> 4;
  for (int mt = blockIdx.x; mt < mtiles; mt += gridDim.x) {
    // Cooperative 128-bit staging of the 16xK A strip.
    for (int idx = threadIdx.x; idx < 16 * (K / 4); idx += blockDim.x) {
      const int r  = idx >> 5;     // 32 float4 per row
      const int c4 = idx & 31;
      int rg = mt * 16 + r;
      if (rg >= M) rg = M - 1;     // harmless clamp (M is a multiple of 16 here)
      const float4 v =
          *reinterpret_cast<const float4*>(&A[(size_t)rg * K + c4 * 4]);
      *reinterpret_cast<float4*>(&Alds[r * LDW + c4 * 4]) = v;
    }
    __syncthreads();

    f32x8 acc = {};
#pragma unroll
    for (int ks = 0; ks < K / 4; ++ks) {
      const f32x2 a = *reinterpret_cast<const f32x2*>(
          &Alds[row_in * LDW + ks * 4 + kb]);
      acc = __builtin_amdgcn_wmma_f32_16x16x4_f32(
          false, a, false, bf[ks], (short)0, acc, false, false);
    }
    __syncthreads();

#pragma unroll
    for (int j = 0; j < 8; ++j) {
      const int rg = mt * 16 + j + ((lane >> 4) << 3);
      if (rg < M) Out[(size_t)rg * NCOL + col] = acc[j];
    }
  }
}

// ---------------------------------------------------------------------------
// Graph-normalization helper kernels
// ---------------------------------------------------------------------------
__global__ void init_deg(float* deg, int n) {
  int i = blockIdx.x * blockDim.x + threadIdx.x;
  if (i < n) deg[i] = 1.0f;  // self-loop weight
}

__global__ void accum_deg(const int* __restrict__ dst,
                          const float* __restrict__ w, float* deg, int E) {
  int e = blockIdx.x * blockDim.x + threadIdx.x;
  if (e < E) atomicAdd(&deg[dst[e]], w[e]);
}

__global__ void make_dinv(float* deg, int n) {
  int i = blockIdx.x * blockDim.x + threadIdx.x;
  if (i < n) {
    float d = deg[i];
    deg[i] = (d > 0.0f) ? rsqrtf(d) : 0.0f;
  }
}

__global__ void make_norm(const int* __restrict__ src,
                          const int* __restrict__ dst,
                          const float* __restrict__ w,
                          const float* __restrict__ dinv,
                          float* __restrict__ nrm, int E) {
  int e = blockIdx.x * blockDim.x + threadIdx.x;
  if (e < E) nrm[e] = dinv[src[e]] * w[e] * dinv[dst[e]];
}

// agg[i,:] = dinv[i]^2 * feat[i,:]   (self-loop contribution; also inits buffer)
template <int C>
__global__ void init_selfloop(const float* __restrict__ feat,
                              const float* __restrict__ dinv,
                              float* __restrict__ agg, int n) {
  constexpr int SH = (C == 128) ? 7 : 6;
  size_t idx = (size_t)blockIdx.x * blockDim.x + threadIdx.x;
  if (idx < (size_t)n * C) {
    const float di = dinv[idx >> SH];
    agg[idx] = di * di * feat[idx];
  }
}

// agg[d,:] += norm[e] * feat[s,:]; C/4 lanes per edge -> coalesced gather/atomics
template <int C>
__global__ void scatter_edges(const float* __restrict__ feat,
                              const float* __restrict__ nrm,
                              const int* __restrict__ src,
                              const int* __restrict__ dst,
                              float* __restrict__ agg, int E) {
  constexpr int G   = C / 4;
  constexpr int GSH = (C == 128) ? 5 : 4;
  unsigned idx = blockIdx.x * blockDim.x + threadIdx.x;
  if (idx >= (unsigned)E * G) return;
  const int e = idx >> GSH;
  const int g = idx & (G - 1);
  const int s = src[e];
  const int d = dst[e];
  const float nm = nrm[e];
  const float4 v =
      *reinterpret_cast<const float4*>(&feat[(size_t)s * C + g * 4]);
  float* p = &agg[(size_t)d * C + g * 4];
  atomicAdd(p + 0, nm * v.x);
  atomicAdd(p + 1, nm * v.y);
  atomicAdd(p + 2, nm * v.z);
  atomicAdd(p + 3, nm * v.w);
}

template <int C>
__global__ void elu_bias(const float* __restrict__ b, float* __restrict__ h,
                         size_t total) {
  size_t idx = (size_t)blockIdx.x * blockDim.x + threadIdx.x;
  if (idx < total) {
    const float z = h[idx] + b[idx & (C - 1)];
    h[idx] = (z > 0.0f) ? z : expm1f(z);
  }
}

template <int C>
__global__ void softplus_bias(const float* __restrict__ b, float* __restrict__ o,
                              size_t total) {
  size_t idx = (size_t)blockIdx.x * blockDim.x + threadIdx.x;
  if (idx < total) {
    const float z  = o[idx] + b[idx & (C - 1)];
    const float sp = (z > 0.0f) ? z + log1pf(expf(-z)) : log1pf(expf(z));
    o[idx] = sp + 1e-4f;
  }
}

// ---------------------------------------------------------------------------
// Orchestration
// ---------------------------------------------------------------------------
extern "C" void kernel_launch(void* const* d_in, const int* in_sizes, int n_in,
                              void* d_out, int out_size, void* d_ws,
                              size_t ws_size, hipStream_t stream) {
  const float* x  = (const float*)d_in[0];
  const int*   ei = (const int*)d_in[1];   // [2, E]: src row then dst row
  const float* ew = (const float*)d_in[2];
  const float* W1 = (const float*)d_in[3];
  const float* b1 = (const float*)d_in[4];
  const float* W2 = (const float*)d_in[5];
  const float* b2 = (const float*)d_in[6];

  const int N = in_sizes[0] / CIN;
  const int E = in_sizes[2];
  const int* src = ei;
  const int* dst = ei + E;

  // Workspace layout (floats): dinv[N] | norm[E] | xw[N*128] | h/agg1[N*128]
  float* ws   = (float*)d_ws;
  float* dinv = ws;
  float* nrm  = dinv + N;
  float* xw   = nrm + E;
  float* hbuf = xw + (size_t)N * CHID;
  float* hw   = xw;                 // reuse: xw dead after layer-1 activation
  float* outf = (float*)d_out;      // agg2 accumulates directly into d_out

  const int T = 256;
  const int mtiles = (N + 15) >> 4;

  // --- normalization terms ---
  init_deg<<<(N + T - 1) / T, T, 0, stream>>>(dinv, N);
  accum_deg<<<(E + T - 1) / T, T, 0, stream>>>(dst, ew, dinv, E);
  make_dinv<<<(N + T - 1) / T, T, 0, stream>>>(dinv, N);
  make_norm<<<(E + T - 1) / T, T, 0, stream>>>(src, dst, ew, dinv, nrm, E);

  // --- layer 1: xw = x @ W1 ; aggregate ; ELU(+b1) ---
  gemm_wmma_f32<CHID><<<mtiles, 2 * CHID, 0, stream>>>(x, W1, xw, N);
  {
    size_t tot = (size_t)N * CHID;
    init_selfloop<CHID><<<(unsigned)((tot + T - 1) / T), T, 0, stream>>>(
        xw, dinv, hbuf, N);
    unsigned work = (unsigned)E * (CHID / 4);
    scatter_edges<CHID><<<(work + T - 1) / T, T, 0, stream>>>(
        xw, nrm, src, dst, hbuf, E);
    elu_bias<CHID><<<(unsigned)((tot + T - 1) / T), T, 0, stream>>>(b1, hbuf,
                                                                    tot);
  }

  // --- layer 2: hw = h @ W2 ; aggregate into d_out ; softplus(+b2)+1e-4 ---
  gemm_wmma_f32<COUT><<<mtiles, 2 * COUT, 0, stream>>>(hbuf, W2, hw, N);
  {
    size_t tot = (size_t)N * COUT;
    init_selfloop<COUT><<<(unsigned)((tot + T - 1) / T), T, 0, stream>>>(
        hw, dinv, outf, N);
    unsigned work = (unsigned)E * (COUT / 4);
    scatter_edges<COUT><<<(work + T - 1) / T, T, 0, stream>>>(
        hw, nrm, src, dst, outf, E);
    softplus_bias<COUT><<<(unsigned)((tot + T - 1) / T), T, 0, stream>>>(
        b2, outf, tot);
  }
}